// ManualMaxPool2d_37031208026365
// MI455X (gfx1250) — compile-verified
//
#include <hip/hip_runtime.h>
#include <math.h>

// Problem constants (from the reference): NCHW = 32,64,224,224; k=3,s=2,p=1.
#define N_      32
#define C_      64
#define H_      224
#define W_      224
#define OH_     112
#define OW_     112

#define TILE_OH        16                     // output rows per block
#define TILES_PER_PL   (OH_ / TILE_OH)        // 7
#define IN_ROWS        (2 * TILE_OH + 1)      // 33 input rows per tile
#define LDS_STRIDE     232                    // words per LDS row; data @ word 4..227
#define NCHUNK         (W_ / 4)               // 56 float4 chunks per input row
#define TOTAL_CHUNKS   (IN_ROWS * NCHUNK)     // 1848
#define BLOCK_THREADS  256                    // 8 wave32's
#define OUT_PER_THREAD ((TILE_OH * OW_) / BLOCK_THREADS)  // 7 exactly

// ---- CDNA5 async global->LDS path (ASYNCcnt), with compile-safe fallbacks ----
#if defined(__has_builtin)
#if __has_builtin(__builtin_amdgcn_global_load_async_to_lds_b128)
#define HAVE_ASYNC_LDS 1
#endif
#endif
#ifndef HAVE_ASYNC_LDS
#define HAVE_ASYNC_LDS 0
#endif

#if defined(__has_builtin)
#if __has_builtin(__builtin_amdgcn_s_wait_asynccnt)
#define WAIT_ASYNCCNT0() __builtin_amdgcn_s_wait_asynccnt(0)
#endif
#endif
#ifndef WAIT_ASYNCCNT0
#define WAIT_ASYNCCNT0() asm volatile("s_wait_asynccnt 0" ::: "memory")
#endif

typedef int v4i __attribute__((ext_vector_type(4)));

__global__ __launch_bounds__(BLOCK_THREADS)
void maxpool3x3s2p1_kernel(const float* __restrict__ x, float* __restrict__ out) {
    __shared__ __align__(16) float lds[IN_ROWS * LDS_STRIDE];

    const int t     = threadIdx.x;
    const int blk   = blockIdx.x;
    const int plane = blk / TILES_PER_PL;          // n*C + c
    const int tile  = blk % TILES_PER_PL;
    const int oh0   = tile * TILE_OH;

    const float* __restrict__ src = x + (size_t)plane * (H_ * W_);
    const float NEG_INF = -__builtin_inff();

    // Left/right -inf padding columns (word 3 and word 228 of each LDS row).
    for (int i = t; i < IN_ROWS * 2; i += BLOCK_THREADS) {
        const int r    = i >> 1;
        const int cpad = (i & 1) ? 228 : 3;
        lds[r * LDS_STRIDE + cpad] = NEG_INF;
    }

    // Stream 33 input rows into LDS, each element read exactly once.
    const int ihBase = 2 * oh0 - 1;
    for (int idx = t; idx < TOTAL_CHUNKS; idx += BLOCK_THREADS) {
        const int r  = idx / NCHUNK;
        const int c4 = idx % NCHUNK;
        const int ih = ihBase + r;
        float* dstp = &lds[r * LDS_STRIDE + 4 + 4 * c4];
        if (ih >= 0 && ih < H_) {
            const float* gp = src + (size_t)ih * W_ + 4 * c4;
#if HAVE_ASYNC_LDS
            // 16B async DMA global -> LDS; tracked with ASYNCcnt; no VGPR staging.
            __builtin_amdgcn_global_load_async_to_lds_b128(
                (v4i*)gp, (v4i*)dstp, 0, 0);
#else
            *(float4*)dstp = *(const float4*)gp;
#endif
        } else {
            // Only ih == -1 on the first tile: torch-style -inf padding row.
            float4 v; v.x = v.y = v.z = v.w = NEG_INF;
            *(float4*)dstp = v;
        }
    }

#if HAVE_ASYNC_LDS
    WAIT_ASYNCCNT0();      // this wave's async LDS writes are complete
#endif
    __syncthreads();       // all waves' fills visible

    // Each thread computes 7 outputs; LDS word = row*232 + 3 + 2*ow + kw.
    float* __restrict__ outp = out + (size_t)plane * (OH_ * OW_) + (size_t)oh0 * OW_;
#pragma unroll
    for (int k = 0; k < OUT_PER_THREAD; ++k) {
        const int flat = t + k * BLOCK_THREADS;    // 0 .. 1791
        const int ohl  = flat / OW_;               // 0 .. 15
        const int ow   = flat % OW_;

        const float* r0 = &lds[(2 * ohl) * LDS_STRIDE + 3 + 2 * ow];
        const float* r1 = r0 + LDS_STRIDE;
        const float* r2 = r1 + LDS_STRIDE;

        float m = r0[0];
        m = fmaxf(m, r0[1]); m = fmaxf(m, r0[2]);
        m = fmaxf(m, r1[0]); m = fmaxf(m, r1[1]); m = fmaxf(m, r1[2]);
        m = fmaxf(m, r2[0]); m = fmaxf(m, r2[1]); m = fmaxf(m, r2[2]);

        outp[(size_t)ohl * OW_ + ow] = m;          // coalesced per wave
    }
}

extern "C" void kernel_launch(void* const* d_in, const int* in_sizes, int n_in,
                              void* d_out, int out_size, void* d_ws, size_t ws_size,
                              hipStream_t stream) {
    (void)in_sizes; (void)n_in; (void)d_ws; (void)ws_size; (void)out_size;
    const float* x = (const float*)d_in[0];
    float* out = (float*)d_out;

    const int planes = N_ * C_;                    // 2048
    dim3 grid(planes * TILES_PER_PL);              // 14336 blocks
    dim3 block(BLOCK_THREADS);
    maxpool3x3s2p1_kernel<<<grid, block, 0, stream>>>(x, out);
}